// GNNPolicy_27857157882445
// MI455X (gfx1250) — compile-verified
//
#include <hip/hip_runtime.h>
#include <hip/hip_bf16.h>

typedef float v2f __attribute__((ext_vector_type(2)));
typedef float v8f __attribute__((ext_vector_type(8)));

#define LN_EPS 1e-5f

__device__ __forceinline__ v8f wmma_f32_k4(v2f a, v2f b, v8f c) {
  // D = A(16x4 f32) * B(4x16 f32) + C(16x16 f32), full fp32 precision
  return __builtin_amdgcn_wmma_f32_16x16x4_f32(false, a, false, b, (short)0, c,
                                               false, false);
}

// ---------------------------------------------------------------- utilities
__global__ void k_zero(float* __restrict__ p, long n) {
  long i = (long)blockIdx.x * blockDim.x + threadIdx.x;
  if (i < n) p[i] = 0.f;
}

// --------------------------------------------------- column stats (BN / GN)
// accumulate sum and sumsq per column into gs1/gs2 (pre-zeroed)
__global__ void k_colstats(const float* __restrict__ X, long N, int F,
                           float* __restrict__ gs1, float* __restrict__ gs2) {
  __shared__ float r1[256];
  __shared__ float r2[256];
  int f = threadIdx.x % F;
  int sub = threadIdx.x / F;
  int nsub = blockDim.x / F;
  float s1 = 0.f, s2 = 0.f;
  if (sub < nsub) {
    for (long r = (long)blockIdx.x * nsub + sub; r < N;
         r += (long)gridDim.x * nsub) {
      float v = X[r * F + f];
      s1 += v;
      s2 += v * v;
    }
  }
  r1[threadIdx.x] = s1;
  r2[threadIdx.x] = s2;
  __syncthreads();
  if (threadIdx.x < (unsigned)F) {
    float t1 = 0.f, t2 = 0.f;
    for (int j = 0; j < nsub; j++) {
      t1 += r1[j * F + threadIdx.x];
      t2 += r2[j * F + threadIdx.x];
    }
    atomicAdd(&gs1[threadIdx.x], t1);
    atomicAdd(&gs2[threadIdx.x], t2);
  }
}

// finalize: mu = s1/N ; var = E[x^2] - 2*ms*mu^2 + ms^2*mu^2  (ms==1 -> BN)
__global__ void k_statfin(const float* __restrict__ gs1,
                          const float* __restrict__ gs2, long N, int F,
                          const float* __restrict__ ms, float* __restrict__ mu,
                          float* __restrict__ rinv) {
  int f = threadIdx.x;
  if (f >= F) return;
  float m = gs1[f] / (float)N;
  float ex2 = gs2[f] / (float)N;
  float s = ms ? ms[f] : 1.f;
  float var = ex2 - 2.f * s * m * m + s * s * m * m;
  mu[f] = m;
  rinv[f] = rsqrtf(var + LN_EPS);
}

// y = (x - ms*mu) * rinv * g + b   (BN when ms==nullptr, GraphNorm otherwise)
__global__ void k_normapply(const float* __restrict__ X, float* __restrict__ Y,
                            long N, int F, const float* __restrict__ mu,
                            const float* __restrict__ rinv,
                            const float* __restrict__ g,
                            const float* __restrict__ b,
                            const float* __restrict__ ms) {
  long i = (long)blockIdx.x * blockDim.x + threadIdx.x;
  if (i >= N * (long)F) return;
  int f = (int)(i % F);
  float s = ms ? ms[f] : 1.f;
  Y[i] = (X[i] - s * mu[f]) * rinv[f] * g[f] + b[f];
}

// ------------------------------------------------ tiny-K input layers (K<=19)
__global__ void k_tinygemm(const float* __restrict__ A,
                           const float* __restrict__ W,
                           const float* __restrict__ bias,
                           float* __restrict__ Y, long M, int K, int relu) {
  __shared__ float Wl[2048];  // K*64 <= 19*64
  for (int i = threadIdx.x; i < K * 64; i += blockDim.x) Wl[i] = W[i];
  __syncthreads();
  long gid = (long)blockIdx.x * blockDim.x + threadIdx.x;
  if (gid >= M * 64) return;
  long m = gid >> 6;
  int n = (int)(gid & 63);
  float s = bias ? bias[n] : 0.f;
  const float* a = A + m * K;
  for (int k = 0; k < K; k++) s += a[k] * Wl[k * 64 + n];
  if (relu) s = fmaxf(s, 0.f);
  Y[gid] = s;
}

// ----------------------------------------------- generic WMMA GEMM (K<=128)
// Y[M,N] = optrelu(A[M,K] @ W[K,N] + bias). M tiles of 16 rows per block,
// one 16x16 C tile per wave. A tile staged in LDS; B streamed from L2.
__global__ void k_wmma_gemm(const float* __restrict__ A,
                            const float* __restrict__ W,
                            const float* __restrict__ bias,
                            float* __restrict__ Y, long M, int N, int K,
                            int relu) {
  __shared__ float As[16 * 132];
  const int pitch = K + 4;
  long m0 = (long)blockIdx.x * 16;
  for (int idx = threadIdx.x; idx < 16 * K; idx += blockDim.x) {
    int r = idx / K;
    int k = idx - r * K;
    long gm = m0 + r;
    if (gm >= M) gm = M - 1;
    As[r * pitch + k] = A[gm * K + k];
  }
  __syncthreads();
  int lane = threadIdx.x & 31;
  int wave = threadIdx.x >> 5;
  int li = lane & 15;
  int hi = lane >> 4;
  int wpb = blockDim.x >> 5;
  int nt = blockIdx.y * wpb + wave;
  if (nt * 16 >= N) return;
  int n0 = nt * 16;
  v8f acc = {0.f, 0.f, 0.f, 0.f, 0.f, 0.f, 0.f, 0.f};
  for (int k0 = 0; k0 < K; k0 += 4) {
    int kk = k0 + 2 * hi;
    v2f va, vb;
    va.x = As[li * pitch + kk];
    va.y = As[li * pitch + kk + 1];
    vb.x = W[(long)kk * N + n0 + li];
    vb.y = W[(long)(kk + 1) * N + n0 + li];
    acc = wmma_f32_k4(va, vb, acc);
  }
  for (int r = 0; r < 8; r++) {
    long gr = m0 + r + 8 * hi;
    if (gr >= M) continue;
    int gc = n0 + li;
    float v = acc[r] + (bias ? bias[gc] : 0.f);
    if (relu) v = fmaxf(v, 0.f);
    Y[gr * (long)N + gc] = v;
  }
}

// --------------------------- fused edge stage of biconv (the hot kernel)
// per edge: z = ti[dst] + tj[src] + en*We ; h = relu(LN(z)) @ Wf + bf ;
// agg[dst] += h.   16 edges per wave (LN via shfl + LDS tile), WMMA GEMM,
// scatter via global f32 atomics.
__global__ void k_edge_biconv(
    const float* __restrict__ ti, const float* __restrict__ tj,
    const float* __restrict__ en, const float* __restrict__ We,
    const int* __restrict__ srcI, const int* __restrict__ dstI,
    const float* __restrict__ lng, const float* __restrict__ lnb,
    const float* __restrict__ Wf, const float* __restrict__ bfp,
    float* __restrict__ agg, long E) {
  __shared__ float WfL[64 * 64];
  __shared__ float WeL[64], lngL[64], lnbL[64], bfL[64];
  __shared__ float Z[8][16 * 65];  // stride 65 -> conflict-free tile
  __shared__ int dstL[8][16];
  for (int i = threadIdx.x; i < 4096; i += blockDim.x) WfL[i] = Wf[i];
  if (threadIdx.x < 64) {
    WeL[threadIdx.x] = We[threadIdx.x];
    lngL[threadIdx.x] = lng[threadIdx.x];
    lnbL[threadIdx.x] = lnb[threadIdx.x];
    bfL[threadIdx.x] = bfp[threadIdx.x];
  }
  __syncthreads();

  int lane = threadIdx.x & 31;
  int w = threadIdx.x >> 5;
  int li = lane & 15;
  int hi = lane >> 4;
  long e = (long)blockIdx.x * 128 + w * 16 + li;
  bool valid = e < E;
  long ec = valid ? e : (E - 1);
  int s = srcI[ec];
  int d = dstI[ec];
  float ev = en[ec];
  if (hi == 0) dstL[w][li] = valid ? d : -1;

  const float* tiR = ti + (long)d * 64 + hi * 32;
  const float* tjR = tj + (long)s * 64 + hi * 32;
  float zr[32];
  float sum = 0.f, sq = 0.f;
  for (int j = 0; j < 32; j++) {
    float z = tiR[j] + tjR[j] + ev * WeL[hi * 32 + j];
    zr[j] = z;
    sum += z;
    sq += z * z;
  }
  sum += __shfl_xor(sum, 16);
  sq += __shfl_xor(sq, 16);
  float mean = sum * (1.f / 64.f);
  float rstd = rsqrtf(sq * (1.f / 64.f) - mean * mean + LN_EPS);
  for (int j = 0; j < 32; j++) {
    int f = hi * 32 + j;
    float z = (zr[j] - mean) * rstd * lngL[f] + lnbL[f];
    Z[w][li * 65 + f] = fmaxf(z, 0.f);
  }
  __syncthreads();

  for (int ntile = 0; ntile < 4; ntile++) {
    v8f acc = {0.f, 0.f, 0.f, 0.f, 0.f, 0.f, 0.f, 0.f};
    for (int k0 = 0; k0 < 64; k0 += 4) {
      int kk = k0 + 2 * hi;
      v2f va, vb;
      va.x = Z[w][li * 65 + kk];
      va.y = Z[w][li * 65 + kk + 1];
      vb.x = WfL[kk * 64 + ntile * 16 + li];
      vb.y = WfL[(kk + 1) * 64 + ntile * 16 + li];
      acc = wmma_f32_k4(va, vb, acc);
    }
    for (int r = 0; r < 8; r++) {
      int row = r + 8 * hi;
      int dd = dstL[w][row];
      if (dd >= 0) {
        int col = ntile * 16 + li;
        atomicAdd(&agg[(long)dd * 64 + col], acc[r] + bfL[col]);
      }
    }
  }
}

// --------------------------------------- row LayerNorm over 64 features
__global__ void k_ln64(const float* __restrict__ X, const float* __restrict__ g,
                       const float* __restrict__ b, float* __restrict__ Y,
                       int ystride, long N) {
  int lane = threadIdx.x & 31;
  int w = threadIdx.x >> 5;
  long row = (long)blockIdx.x * (blockDim.x >> 5) + w;
  if (row >= N) return;
  float x0 = X[row * 64 + lane];
  float x1 = X[row * 64 + 32 + lane];
  float s = x0 + x1, q = x0 * x0 + x1 * x1;
  for (int off = 16; off; off >>= 1) {
    s += __shfl_xor(s, off);
    q += __shfl_xor(q, off);
  }
  float m = s * (1.f / 64.f);
  float rstd = rsqrtf(q * (1.f / 64.f) - m * m + LN_EPS);
  Y[row * (long)ystride + lane] = (x0 - m) * rstd * g[lane] + b[lane];
  Y[row * (long)ystride + 32 + lane] =
      (x1 - m) * rstd * g[32 + lane] + b[32 + lane];
}

__global__ void k_copy64(const float* __restrict__ X, float* __restrict__ Y,
                         int ystride, int yoff, long N) {
  long i = (long)blockIdx.x * blockDim.x + threadIdx.x;
  if (i >= N * 64) return;
  long r = i >> 6;
  int f = (int)(i & 63);
  Y[r * (long)ystride + yoff + f] = X[i];
}

// ------------------------------------------------ consnorm (degree norm)
__global__ void k_cn_scatter1(const float* __restrict__ v,
                              const int* __restrict__ vIdx,
                              const int* __restrict__ cIdx,
                              const int* __restrict__ ncons,
                              float* __restrict__ sc, float* __restrict__ cc,
                              long E) {
  long gid = (long)blockIdx.x * blockDim.x + threadIdx.x;
  if (gid >= E * 16) return;
  long e = gid >> 4;
  int fg = (int)(gid & 15) * 4;
  int last = ncons[0] - 1;
  int ci = cIdx[e];
  if (ci == last) return;
  int vi = vIdx[e];
  if (fg == 0) atomicAdd(&cc[ci], 1.f);
  const float* vr = v + (long)vi * 64 + fg;
  float* so = sc + (long)ci * 64 + fg;
  for (int j = 0; j < 4; j++) atomicAdd(&so[j], vr[j]);
}

__global__ void k_cn_scatter2(const float* __restrict__ sc,
                              const float* __restrict__ cc,
                              const int* __restrict__ vIdx,
                              const int* __restrict__ cIdx,
                              const int* __restrict__ ncons,
                              float* __restrict__ sv, float* __restrict__ cv,
                              long E) {
  long gid = (long)blockIdx.x * blockDim.x + threadIdx.x;
  if (gid >= E * 16) return;
  long e = gid >> 4;
  int fg = (int)(gid & 15) * 4;
  int last = ncons[0] - 1;
  int ci = cIdx[e];
  if (ci == last) return;
  int vi = vIdx[e];
  if (fg == 0) atomicAdd(&cv[vi], 1.f);
  float inv = 1.f / fmaxf(cc[ci], 1.f);
  const float* sr = sc + (long)ci * 64 + fg;
  float* so = sv + (long)vi * 64 + fg;
  for (int j = 0; j < 4; j++) atomicAdd(&so[j], sr[j] * inv);
}

__global__ void k_cn_update(float* __restrict__ v, const float* __restrict__ sv,
                            const float* __restrict__ cv,
                            const float* __restrict__ beta, long N) {
  long i = (long)blockIdx.x * blockDim.x + threadIdx.x;
  if (i >= N * 64) return;
  long n = i >> 6;
  v[i] -= beta[0] * sv[i] / fmaxf(cv[n], 1.f);
}

// ------------------- fused output MLP: 192 -> 512 -> 128 -> 1, per 16 nodes
__global__ void k_out_mlp(const float* __restrict__ vcat,
                          const float* __restrict__ W1,
                          const float* __restrict__ b1,
                          const float* __restrict__ W2,
                          const float* __restrict__ b2,
                          const float* __restrict__ W3,
                          const float* __restrict__ b3, float* __restrict__ out,
                          long N) {
  __shared__ float A0[16 * 196];
  __shared__ float H1[16 * 516];
  __shared__ float H2[16 * 132];
  long m0 = (long)blockIdx.x * 16;
  for (int idx = threadIdx.x; idx < 16 * 192; idx += blockDim.x) {
    int r = idx / 192;
    int k = idx - r * 192;
    long gm = m0 + r;
    if (gm >= N) gm = N - 1;
    A0[r * 196 + k] = vcat[gm * 192 + k];
  }
  __syncthreads();
  int lane = threadIdx.x & 31;
  int w = threadIdx.x >> 5;
  int li = lane & 15;
  int hi = lane >> 4;

  // stage 1: [16x192] @ [192x512] -> H1 (relu)
  for (int t = 0; t < 4; t++) {
    int n0 = (w + t * 8) * 16;
    v8f acc = {0.f, 0.f, 0.f, 0.f, 0.f, 0.f, 0.f, 0.f};
    for (int k0 = 0; k0 < 192; k0 += 4) {
      int kk = k0 + 2 * hi;
      v2f va, vb;
      va.x = A0[li * 196 + kk];
      va.y = A0[li * 196 + kk + 1];
      vb.x = W1[(long)kk * 512 + n0 + li];
      vb.y = W1[(long)(kk + 1) * 512 + n0 + li];
      acc = wmma_f32_k4(va, vb, acc);
    }
    for (int r = 0; r < 8; r++) {
      int row = r + 8 * hi;
      int col = n0 + li;
      H1[row * 516 + col] = fmaxf(acc[r] + b1[col], 0.f);
    }
  }
  __syncthreads();

  // stage 2: [16x512] @ [512x128] -> H2 (relu)
  {
    int n0 = w * 16;
    v8f acc = {0.f, 0.f, 0.f, 0.f, 0.f, 0.f, 0.f, 0.f};
    for (int k0 = 0; k0 < 512; k0 += 4) {
      int kk = k0 + 2 * hi;
      v2f va, vb;
      va.x = H1[li * 516 + kk];
      va.y = H1[li * 516 + kk + 1];
      vb.x = W2[(long)kk * 128 + n0 + li];
      vb.y = W2[(long)(kk + 1) * 128 + n0 + li];
      acc = wmma_f32_k4(va, vb, acc);
    }
    for (int r = 0; r < 8; r++) {
      int row = r + 8 * hi;
      int col = n0 + li;
      H2[row * 132 + col] = fmaxf(acc[r] + b2[col], 0.f);
    }
  }
  __syncthreads();

  // stage 3: per-row dot with W3[128]
  {
    int row = 2 * w + hi;
    float s = 0.f;
    for (int f = li; f < 128; f += 16) s += H2[row * 132 + f] * W3[f];
    for (int off = 8; off; off >>= 1) s += __shfl_xor(s, off, 16);
    long gm = m0 + row;
    if (li == 0 && gm < N) out[gm] = s + b3[0];
  }
}

// =========================================================== orchestration
extern "C" void kernel_launch(void* const* d_in, const int* in_sizes, int n_in,
                              void* d_out, int out_size, void* d_ws,
                              size_t ws_size, hipStream_t stream) {
  const float* cons = (const float*)d_in[0];
  const int* eidx = (const int*)d_in[1];
  const float* efeat = (const float*)d_in[2];
  const float* varf = (const float*)d_in[3];
  const int* nconsPtr = (const int*)d_in[4];

  const long NC = in_sizes[0] / 5;
  const long E = in_sizes[2];
  const long NV = in_sizes[3] / 19;
  const int* consRow = eidx;      // edge_indices[0] : constraint index
  const int* varRow = eidx + E;   // edge_indices[1] : variable index

  // params: d_in[5..], insertion order of the setup_inputs() dicts
  auto P = [&](int i) -> const float* { return (const float*)d_in[5 + i]; };
  // 0..13: input embed params. v2c[i] base 14+i*14, c2v[i] base 42+i*14,
  // gn_c[i] 70+i*3, gn_v[i] 76+i*3, cn_beta[i] 82+i, out head 84..89
  // (these are offsets relative to param index 0 == d_in[5])

  // -------- workspace layout (floats)
  float* Wp = (float*)d_ws;
  size_t off = 0;
  auto alloc = [&](size_t n) {
    float* p = Wp + off;
    off += n;
    return p;
  };
  float* gs1 = alloc(64);
  float* gs2 = alloc(64);
  float* mu = alloc(64);
  float* rinv = alloc(64);
  float* cc = alloc(NC);
  float* cv = alloc(NV);
  size_t xnsz = (size_t)(NC * 5 > NV * 19 ? NC * 5 : NV * 19);
  float* xn = alloc(xnsz);
  float* c_buf = alloc((size_t)NC * 64);
  float* v_buf = alloc((size_t)NV * 64);
  float* vcat = alloc((size_t)NV * 192);
  float* e_n = alloc((size_t)E);
  float* ti = alloc((size_t)NV * 64);
  float* tj = alloc((size_t)NV * 64);
  float* agg = alloc((size_t)NV * 64);
  float* cat = alloc((size_t)NV * 128);
  float* tmpA = alloc((size_t)NV * 64);
  (void)ws_size;

  auto zero = [&](float* p, long n) {
    k_zero<<<dim3((unsigned)((n + 255) / 256)), dim3(256), 0, stream>>>(p, n);
  };
  auto stats = [&](const float* X, long N, int F, const float* ms) {
    zero(gs1, 64);
    zero(gs2, 64);
    k_colstats<<<dim3(256), dim3(256), 0, stream>>>(X, N, F, gs1, gs2);
    k_statfin<<<dim3(1), dim3(64), 0, stream>>>(gs1, gs2, N, F, ms, mu, rinv);
  };
  auto apply = [&](const float* X, float* Y, long N, int F, const float* g,
                   const float* b, const float* ms) {
    long n = N * F;
    k_normapply<<<dim3((unsigned)((n + 255) / 256)), dim3(256), 0, stream>>>(
        X, Y, N, F, mu, rinv, g, b, ms);
  };
  auto gemm = [&](const float* A, const float* Wm, const float* bias, float* Y,
                  long M, int N, int K, int relu) {
    int wpb = N / 16;
    if (wpb > 8) wpb = 8;
    dim3 grid((unsigned)((M + 15) / 16), (unsigned)((N / 16 + wpb - 1) / wpb));
    k_wmma_gemm<<<grid, dim3((unsigned)(wpb * 32)), 0, stream>>>(A, Wm, bias, Y,
                                                                 M, N, K, relu);
  };
  auto copy64 = [&](const float* X, float* Y, int ystride, int yoff, long N) {
    long n = N * 64;
    k_copy64<<<dim3((unsigned)((n + 255) / 256)), dim3(256), 0, stream>>>(
        X, Y, ystride, yoff, N);
  };

  // ---------------- input embeddings
  // constraints: BN(5) -> 5x64 relu -> 64x64 relu
  stats(cons, NC, 5, nullptr);
  apply(cons, xn, NC, 5, P(0), P(1), nullptr);
  k_tinygemm<<<dim3((unsigned)((NC * 64 + 255) / 256)), dim3(256), 0, stream>>>(
      xn, P(2), P(3), tmpA, NC, 5, 1);
  gemm(tmpA, P(4), P(5), c_buf, NC, 64, 64, 1);
  // edges: BN(1)
  stats(efeat, E, 1, nullptr);
  apply(efeat, e_n, E, 1, P(6), P(7), nullptr);
  // variables: BN(19) -> 19x64 relu -> 64x64 relu
  stats(varf, NV, 19, nullptr);
  apply(varf, xn, NV, 19, P(8), P(9), nullptr);
  k_tinygemm<<<dim3((unsigned)((NV * 64 + 255) / 256)), dim3(256), 0, stream>>>(
      xn, P(10), P(11), tmpA, NV, 19, 1);
  gemm(tmpA, P(12), P(13), v_buf, NV, 64, 64, 1);
  copy64(v_buf, vcat, 192, 0, NV);  // vouts[0]

  const unsigned egrid = (unsigned)((E + 127) / 128);
  for (int i = 0; i < 2; i++) {
    // ---- v2c: left=v (src=varRow), right=c (dst=consRow)
    int b = 14 + i * 14;
    gemm(c_buf, P(b + 0), P(b + 1), ti, NC, 64, 64, 0);   // feature_left
    gemm(v_buf, P(b + 3), nullptr, tj, NV, 64, 64, 0);    // feature_right
    zero(agg, NC * 64);
    k_edge_biconv<<<dim3(egrid), dim3(256), 0, stream>>>(
        ti, tj, e_n, P(b + 2), varRow, consRow, P(b + 4), P(b + 5), P(b + 6),
        P(b + 7), agg, E);
    k_ln64<<<dim3((unsigned)((NC + 7) / 8)), dim3(256), 0, stream>>>(
        agg, P(b + 8), P(b + 9), cat, 128, NC);
    copy64(c_buf, cat, 128, 64, NC);
    gemm(cat, P(b + 10), P(b + 11), tmpA, NC, 64, 128, 1);
    gemm(tmpA, P(b + 12), P(b + 13), c_buf, NC, 64, 64, 0);

    // ---- c2v: left=c (src=consRow), right=v (dst=varRow)
    b = 42 + i * 14;
    gemm(v_buf, P(b + 0), P(b + 1), ti, NV, 64, 64, 0);
    gemm(c_buf, P(b + 3), nullptr, tj, NC, 64, 64, 0);
    zero(agg, NV * 64);
    k_edge_biconv<<<dim3(egrid), dim3(256), 0, stream>>>(
        ti, tj, e_n, P(b + 2), consRow, varRow, P(b + 4), P(b + 5), P(b + 6),
        P(b + 7), agg, E);
    k_ln64<<<dim3((unsigned)((NV + 7) / 8)), dim3(256), 0, stream>>>(
        agg, P(b + 8), P(b + 9), cat, 128, NV);
    copy64(v_buf, cat, 128, 64, NV);
    gemm(cat, P(b + 10), P(b + 11), tmpA, NV, 64, 128, 1);
    gemm(tmpA, P(b + 12), P(b + 13), v_buf, NV, 64, 64, 0);

    // ---- consnorm (reuse ti as sc, tj as sv)
    zero(ti, NC * 64);
    zero(cc, NC);
    zero(tj, NV * 64);
    zero(cv, NV);
    {
      long n16 = E * 16;
      unsigned g16 = (unsigned)((n16 + 255) / 256);
      k_cn_scatter1<<<dim3(g16), dim3(256), 0, stream>>>(v_buf, varRow, consRow,
                                                         nconsPtr, ti, cc, E);
      k_cn_scatter2<<<dim3(g16), dim3(256), 0, stream>>>(ti, cc, varRow,
                                                         consRow, nconsPtr, tj,
                                                         cv, E);
    }
    k_cn_update<<<dim3((unsigned)((NV * 64 + 255) / 256)), dim3(256), 0,
                  stream>>>(v_buf, tj, cv, P(82 + i), NV);

    // ---- GraphNorm on c and v
    int gb = 70 + i * 3;
    stats(c_buf, NC, 64, P(gb + 2));
    apply(c_buf, c_buf, NC, 64, P(gb + 0), P(gb + 1), P(gb + 2));
    gb = 76 + i * 3;
    stats(v_buf, NV, 64, P(gb + 2));
    apply(v_buf, v_buf, NV, 64, P(gb + 0), P(gb + 1), P(gb + 2));

    copy64(v_buf, vcat, 192, 64 * (i + 1), NV);  // vouts[i+1]
  }

  // ---------------- fused output MLP 192 -> 512 -> 128 -> 1
  k_out_mlp<<<dim3((unsigned)((NV + 15) / 16)), dim3(256), 0, stream>>>(
      vcat, P(84), P(85), P(86), P(87), P(88), P(89), (float*)d_out, NV);
}